// RNNBinaryClassifier_74062416053031
// MI455X (gfx1250) — compile-verified
//
#include <hip/hip_runtime.h>
#include <math.h>

// ---------------- problem constants ----------------
#define BATCH   64
#define T_SEQ   512
#define DIN     2048
#define NH1     256
#define NH2     128
#define NFC     64
#define M1      (BATCH * T_SEQ)   // 32768 rows of the big GEMM

typedef float v2f  __attribute__((ext_vector_type(2)));
typedef float v8f  __attribute__((ext_vector_type(8)));
typedef unsigned int u32x4 __attribute__((ext_vector_type(4)));
typedef int   i32x4 __attribute__((ext_vector_type(4)));
typedef int   i32x8 __attribute__((ext_vector_type(8)));

// Exact fp32 WMMA: D(16x16) = A(16x4) * B(4x16) + C
// A per lane: row = lane%16, K = 2*(lane/16) + {0,1}  -> float2
// B per lane: col = lane%16, K = 2*(lane/16) + {0,1}  -> float2 (B[k][n] = W[n][k])
// C per lane: elem i -> row = i + 8*(lane/16), col = lane%16
__device__ __forceinline__ v8f wmma_f32(v2f a, v2f b, v8f c) {
    return __builtin_amdgcn_wmma_f32_16x16x4_f32(
        /*neg_a=*/false, a, /*neg_b=*/false, b,
        /*c_mod=*/(short)0, c, /*reuse_a=*/false, /*reuse_b=*/false);
}

// Generic pointer to LDS byte offset: flat LDS aperture maps addr[31:0] to LDS.
__device__ __forceinline__ unsigned int lds_off(const void* p) {
    return (unsigned int)(uintptr_t)p;
}

// ---------------------------------------------------------------------
// TDM: async 2D tile load  xp1[:, t, :]  (64 rows x 256 f32, row stride
// T_SEQ*NH1 elements) -> LDS at lds_byte_off.  Tracked by TENSORcnt.
// D# layout per CDNA5 ISA ch.8 (group0 128b, group1 256b; groups 2/3 zero
// since the tensor is 2D).  6-arg builtin form (clang-23 / therock lane).
// ---------------------------------------------------------------------
__device__ __forceinline__ void tdm_load_xp_slice(const float* xp1, int t,
                                                  unsigned int lds_byte_off) {
    const unsigned long long ga =
        (unsigned long long)(uintptr_t)(xp1 + (size_t)t * NH1);
    u32x4 g0;
    g0.x = 1u;                                   // count=1, user descriptor
    g0.y = lds_byte_off;                         // lds_addr [63:32]
    g0.z = (unsigned int)ga;                     // global_addr lo
    g0.w = (unsigned int)(ga >> 32) | (2u << 30); // global_addr hi | type=2
    i32x8 g1;
    g1[0] = (int)(2u << 16);                     // data_size = 4 bytes
    g1[1] = (int)((unsigned)NH1 << 16);          // tensor_dim0 lo16 @ [31:16]
    g1[2] = (int)((unsigned)BATCH << 16);        // t_dim0 hi16 | tensor_dim1 lo16
    g1[3] = (int)((unsigned)NH1 << 16);          // t_dim1 hi16 | tile_dim0 = 256
    g1[4] = (int)BATCH;                          // tile_dim1 = 64, tile_dim2 = 0
    g1[5] = (int)(T_SEQ * NH1);                  // tensor_dim0_stride lo32
    g1[6] = 0;                                   // stride hi16 | dim1_stride lo
    g1[7] = 0;
    const i32x4 z4 = {0, 0, 0, 0};
    const i32x8 z8 = {0, 0, 0, 0, 0, 0, 0, 0};
    __builtin_amdgcn_tensor_load_to_lds(g0, g1, z4, z4, z8, 0);
}

// =====================================================================
// Kernel 1: xp1 = x @ W_ih1^T + (b_ih1 + b_hh1)
//   x: (32768, 2048) fp32, W_ih1: (256, 2048), xp1: (32768, 256)
// Block = 256 threads (8 wave32); wave tile = 16 rows x 128 cols
// (8 wmma accumulators, one A fragment feeds 8 WMMAs).
// Grid = (32768/128, 256/128) = (256, 2)  ->  x HBM traffic = 2 * 256 MB.
// =====================================================================
__global__ __launch_bounds__(256) void gemm1_wmma(
    const float* __restrict__ x,
    const float* __restrict__ Wih1,
    const float* __restrict__ bih1,
    const float* __restrict__ bhh1,
    float* __restrict__ xp1)
{
    const int lane = threadIdx.x & 31;
    const int wave = threadIdx.x >> 5;
    const int lr   = lane & 15;       // row/col within tile
    const int hi   = lane >> 4;       // K-half select
    const int row0 = blockIdx.x * 128 + wave * 16;
    const int col0 = blockIdx.y * 128;

    const float* __restrict__ xrow = x + (size_t)(row0 + lr) * DIN;

    v8f acc[8];
    int ncol[8];
    #pragma unroll
    for (int j = 0; j < 8; ++j) {
        ncol[j] = col0 + j * 16 + lr;
        const float bias = bih1[ncol[j]] + bhh1[ncol[j]];
        #pragma unroll
        for (int i = 0; i < 8; ++i) acc[j][i] = bias;
    }

    #pragma unroll 2
    for (int k = 0; k < DIN; k += 4) {
        const int k0 = k + hi * 2;
        const v2f a = *(const v2f*)(xrow + k0);
        #pragma unroll
        for (int j = 0; j < 8; ++j) {
            const v2f b = *(const v2f*)(Wih1 + (size_t)ncol[j] * DIN + k0);
            acc[j] = wmma_f32(a, b, acc[j]);
        }
    }

    #pragma unroll
    for (int j = 0; j < 8; ++j) {
        #pragma unroll
        for (int i = 0; i < 8; ++i) {
            const int r = row0 + i + hi * 8;
            xp1[(size_t)r * NH1 + ncol[j]] = acc[j][i];
        }
    }
}

// =====================================================================
// Kernel 2: persistent fused recurrence + FC head. One workgroup,
// 1024 threads = 32 wave32 on one WGP.
// LDS (224 KB of the 320 KB/WGP CDNA5 budget):
//   h1 state   64 KB | h2 state 32 KB | xp double-buffer 2 x 64 KB
// Wave 0 drives the Tensor Data Mover: while step t computes, the
// xp1[:, t+1, :] slice is DMA'd into the other LDS buffer (TENSORcnt).
//
// per step t:
//   h1 = tanh(xp_lds + h1 @ W_hh1^T)                        (64x256, K=256)
//   h2 = tanh(b_ih2+b_hh2 + h1 @ W_ih2^T + h2 @ W_hh2^T)    (64x128, K=256+128)
// epilogue: out = sigmoid(relu(h2 @ Wfc1^T + bfc1) @ Wout^T + bout)
// =====================================================================
__global__ __launch_bounds__(1024) void rnn_fused(
    const float* __restrict__ xp1,
    const float* __restrict__ Whh1,
    const float* __restrict__ Wih2,
    const float* __restrict__ Whh2,
    const float* __restrict__ bih2,
    const float* __restrict__ bhh2,
    const float* __restrict__ Wfc1,
    const float* __restrict__ bfc1,
    const float* __restrict__ Wout,
    const float* __restrict__ bout,
    float* __restrict__ out)
{
    extern __shared__ float smem[];
    float* h1  = smem;                          // 64 * 256 floats
    float* h2  = h1 + BATCH * NH1;              // 64 * 128 floats
    float* xb0 = h2 + BATCH * NH2;              // 64 * 256 floats
    float* xb1 = xb0 + BATCH * NH1;             // 64 * 256 floats

    const int tid  = threadIdx.x;
    const int lane = tid & 31;
    const int wave = tid >> 5;        // 0..31
    const int lr   = lane & 15;
    const int hi   = lane >> 4;

    for (int i = tid; i < BATCH * NH1; i += 1024) h1[i] = 0.0f;
    for (int i = tid; i < BATCH * NH2; i += 1024) h2[i] = 0.0f;

    // prime the pipeline: DMA xp1[:,0,:] into buffer 0
    if (wave == 0) tdm_load_xp_slice(xp1, 0, lds_off(xb0));
    __syncthreads();

    // layer-2 tile for this wave: 4 x 8 tile grid over 64x128
    const int rt2 = wave >> 3;
    const int ct2 = wave & 7;
    const int m2  = rt2 * 16 + lr;
    const int n2  = ct2 * 16 + lr;
    const float bias2 = bih2[n2] + bhh2[n2];

    for (int t = 0; t < T_SEQ; ++t) {
        if (wave == 0) {
            __builtin_amdgcn_s_wait_tensorcnt(0);   // xp slice t landed in LDS
            if (t + 1 < T_SEQ)                      // prefetch slice t+1
                tdm_load_xp_slice(xp1, t + 1,
                                  lds_off(((t + 1) & 1) ? xb1 : xb0));
        }
        __syncthreads();                            // slice t visible to all
        const float* __restrict__ xp = (t & 1) ? xb1 : xb0;

        // ---------------- layer 1: 64 tiles over 4x16 grid, 2 per wave ----
        float r1[2][8];
        #pragma unroll
        for (int u = 0; u < 2; ++u) {
            const int tile = wave * 2 + u;
            const int rt = tile >> 4;       // 0..3  (batch rows)
            const int ct = tile & 15;       // 0..15 (H1 cols)
            const int m  = rt * 16 + lr;
            const int n  = ct * 16 + lr;

            v8f c;
            #pragma unroll
            for (int i = 0; i < 8; ++i) {
                const int mi = rt * 16 + i + hi * 8;       // batch index
                c[i] = xp[mi * NH1 + n];
            }

            const float* __restrict__ arow = h1 + m * NH1;      // LDS
            const float* __restrict__ brow = Whh1 + (size_t)n * NH1;
            #pragma unroll 8
            for (int k = 0; k < NH1; k += 4) {
                const int k0 = k + hi * 2;
                const v2f a = *(const v2f*)(arow + k0);
                const v2f b = *(const v2f*)(brow + k0);
                c = wmma_f32(a, b, c);
            }
            #pragma unroll
            for (int i = 0; i < 8; ++i) r1[u][i] = tanhf(c[i]);
        }
        __syncthreads();   // all waves finished reading old h1
        #pragma unroll
        for (int u = 0; u < 2; ++u) {
            const int tile = wave * 2 + u;
            const int rt = tile >> 4;
            const int ct = tile & 15;
            const int n  = ct * 16 + lr;
            #pragma unroll
            for (int i = 0; i < 8; ++i) {
                const int mi = rt * 16 + i + hi * 8;
                h1[mi * NH1 + n] = r1[u][i];
            }
        }
        __syncthreads();   // new h1 visible

        // ---------------- layer 2: 32 tiles over 4x8 grid, 1 per wave -----
        {
            v8f c;
            #pragma unroll
            for (int i = 0; i < 8; ++i) c[i] = bias2;

            // + h1_new @ W_ih2^T   (K = 256)
            const float* __restrict__ arow = h1 + m2 * NH1;
            const float* __restrict__ brow = Wih2 + (size_t)n2 * NH1;
            #pragma unroll 8
            for (int k = 0; k < NH1; k += 4) {
                const int k0 = k + hi * 2;
                const v2f a = *(const v2f*)(arow + k0);
                const v2f b = *(const v2f*)(brow + k0);
                c = wmma_f32(a, b, c);
            }
            // + h2_old @ W_hh2^T   (K = 128)
            const float* __restrict__ arow2 = h2 + m2 * NH2;
            const float* __restrict__ brow2 = Whh2 + (size_t)n2 * NH2;
            #pragma unroll 8
            for (int k = 0; k < NH2; k += 4) {
                const int k0 = k + hi * 2;
                const v2f a = *(const v2f*)(arow2 + k0);
                const v2f b = *(const v2f*)(brow2 + k0);
                c = wmma_f32(a, b, c);
            }

            float r2[8];
            #pragma unroll
            for (int i = 0; i < 8; ++i) r2[i] = tanhf(c[i]);
            __syncthreads();   // all waves finished reading old h2
            #pragma unroll
            for (int i = 0; i < 8; ++i) {
                const int mi = rt2 * 16 + i + hi * 8;
                h2[mi * NH2 + n2] = r2[i];
            }
            __syncthreads();   // new h2 visible
        }
    }

    // ---------------- FC head on final h2 (all LDS-resident) -------------
    float* fcb = h1;   // reuse layer-1 LDS region: 64x64 floats
    #pragma unroll
    for (int j = 0; j < 4; ++j) {
        const int o  = tid * 4 + j;       // 4096 outputs
        const int bi = o >> 6;
        const int n  = o & 63;
        float s = bfc1[n];
        const float* __restrict__ hrow = h2 + bi * NH2;
        const float* __restrict__ wrow = Wfc1 + n * NH2;
        #pragma unroll 4
        for (int k = 0; k < NH2; ++k) s += hrow[k] * wrow[k];
        fcb[bi * 64 + n] = fmaxf(s, 0.0f);
    }
    __syncthreads();
    if (tid < BATCH) {
        float s = bout[0];
        const float* __restrict__ frow = fcb + tid * 64;
        #pragma unroll 4
        for (int k = 0; k < NFC; ++k) s += frow[k] * Wout[k];
        out[tid] = 1.0f / (1.0f + expf(-s));
    }
}

// =====================================================================
// Host launcher
// =====================================================================
extern "C" void kernel_launch(void* const* d_in, const int* in_sizes, int n_in,
                              void* d_out, int out_size, void* d_ws, size_t ws_size,
                              hipStream_t stream) {
    const float* x    = (const float*)d_in[0];
    const float* Wih1 = (const float*)d_in[1];
    const float* Whh1 = (const float*)d_in[2];
    const float* bih1 = (const float*)d_in[3];
    const float* bhh1 = (const float*)d_in[4];
    const float* Wih2 = (const float*)d_in[5];
    const float* Whh2 = (const float*)d_in[6];
    const float* bih2 = (const float*)d_in[7];
    const float* bhh2 = (const float*)d_in[8];
    const float* Wfc1 = (const float*)d_in[9];
    const float* bfc1 = (const float*)d_in[10];
    const float* Wout = (const float*)d_in[11];
    const float* bout = (const float*)d_in[12];
    float* outp = (float*)d_out;

    // workspace: xp1 = (32768, 256) fp32 = 32 MB
    float* xp1 = (float*)d_ws;

    gemm1_wmma<<<dim3(M1 / 128, NH1 / 128), 256, 0, stream>>>(x, Wih1, bih1, bhh1, xp1);

    // LDS: h1 + h2 + two xp slice buffers = 224 KB (CDNA5 WGP: 320 KB)
    const size_t lds_bytes =
        (size_t)(BATCH * NH1 + BATCH * NH2 + 2 * BATCH * NH1) * sizeof(float);
    rnn_fused<<<1, 1024, lds_bytes, stream>>>(xp1, Whh1, Wih2, Whh2, bih2, bhh2,
                                              Wfc1, bfc1, Wout, bout, outp);
    (void)in_sizes; (void)n_in; (void)out_size; (void)ws_size;
}